// ToxLoss_549755814583
// MI455X (gfx1250) — compile-verified
//
#include <hip/hip_runtime.h>
#include <hip/hip_bf16.h>

// ---------------- constants ----------------
constexpr unsigned VOCAB     = 50257u;
constexpr unsigned UNI_PAD   = 50272u;          // multiple of 4 (b128 / 16B granularity)
constexpr unsigned BI_HASH   = 1u << 20;
constexpr unsigned HASH_MULT = 2654435761u;
constexpr int      THREADS   = 512;             // 16 wave32s
constexpr int      ROWS_PER_BLOCK = 8;

typedef __attribute__((ext_vector_type(4))) unsigned uint4v;
typedef __attribute__((ext_vector_type(8))) int      int8v;

// ---------------- prep kernels: fuse tables in workspace ----------------
// ws_uni[v] = ignored ? 0 : (has_uni ? uni+1 : 1.5)   (valid <=> value > 0; contribution = value-1)
__global__ void tox_prep_uni(const float* __restrict__ uni,
                             const int* __restrict__ has_uni,
                             const int* __restrict__ ign,
                             float* __restrict__ ws_uni) {
    unsigned i = blockIdx.x * blockDim.x + threadIdx.x;
    if (i >= UNI_PAD) return;
    float v = 0.0f;
    if (i < VOCAB && ign[i] <= 0)
        v = (has_uni[i] > 0 ? uni[i] : 0.5f) + 1.0f;
    ws_uni[i] = v;
}

// ws_bi[h] = has_bi ? bi : 0
__global__ void tox_prep_bi(const float* __restrict__ bi,
                            const int* __restrict__ has_bi,
                            float* __restrict__ ws_bi) {
    unsigned i = blockIdx.x * blockDim.x + threadIdx.x;
    if (i >= BI_HASH) return;
    ws_bi[i] = (has_bi[i] > 0) ? bi[i] : 0.0f;
}

// ---------------- bigram helper ----------------
template <int BI_MODE> // 1 = fused ws table (1 gather), 0 = raw tables (2 gathers)
__device__ __forceinline__ float bigram_val(int a, int b, float fa, float fb,
                                            const float* __restrict__ bi,
                                            const int* __restrict__ has_bi,
                                            const float* __restrict__ ws_bi) {
    if (fa > 0.0f && fb > 0.0f) {   // ~25% of pairs; EXEC-masked lanes issue no gather
        unsigned h = ((unsigned)a * HASH_MULT + (unsigned)b) & (BI_HASH - 1u);
        if (BI_MODE == 1) return ws_bi[h];
        return (has_bi[h] > 0) ? bi[h] : 0.0f;
    }
    return 0.0f;
}

// ---------------- main kernel ----------------
// UNI_MODE: 2 = single TDM tensor_load_to_lds DMA of the fused table (TENSORcnt)
//           1 = per-lane global_load_async_to_lds_b128 fill          (ASYNCcnt)
//           0 = fuse in-kernel during LDS fill (no workspace needed)
template <int UNI_MODE, int BI_MODE>
__global__ __launch_bounds__(THREADS, 1)
void tox_main(const int* __restrict__ x, int rows, int cols,
              const float* __restrict__ uni, const int* __restrict__ has_uni,
              const int* __restrict__ ign,
              const float* __restrict__ bi, const int* __restrict__ has_bi,
              const float* __restrict__ ws_uni, const float* __restrict__ ws_bi,
              float* __restrict__ out) {
    __shared__ float s_uni[UNI_PAD];   // ~201 KB of the 320 KB WGP LDS
    __shared__ float s_rowsum;

    const unsigned tid = threadIdx.x;

    if (UNI_MODE == 2) {
        // One Tensor Data Mover descriptor moves the whole fused table ws->LDS.
        if (tid < 32u) {               // wave 0 only (EXEC ignored; per-wave op)
            const unsigned lds_addr = (unsigned)(unsigned long long)&s_uni[0];
            const unsigned long long ga = (unsigned long long)ws_uni;
            // D# group 0: count=1 | lds_addr | global_addr[56:0] | type=2
            uint4v g0;
            g0.x = 1u;                                     // count=1, user mode
            g0.y = lds_addr;                               // LDS byte address
            g0.z = (unsigned)(ga & 0xFFFFFFFFull);         // global_addr[31:0]
            g0.w = (unsigned)((ga >> 32) & 0x01FFFFFFull)  // global_addr[56:32]
                 | 0x80000000u;                            // type=2 ("image")
            // D# group 1: data_size=4B, 1-D tile of UNI_PAD elements
            int8v g1;
            g1.s0 = (int)(2u << 16);                       // data_size=2 (4 bytes)
            g1.s1 = (int)((UNI_PAD & 0xFFFFu) << 16);      // tensor_dim0[15:0] @ [63:48]
            g1.s2 = (int)(((UNI_PAD >> 16) & 0xFFFFu)      // tensor_dim0[31:16]
                 | (1u << 16));                            // tensor_dim1 = 1
            g1.s3 = (int)((UNI_PAD & 0xFFFFu) << 16);      // tile_dim0 @ [127:112]
            g1.s4 = 1;                                     // tile_dim1=1, tile_dim2=0
            g1.s5 = (int)UNI_PAD;                          // tensor_dim0_stride[31:0]
            g1.s6 = 0;                                     // stride0[47:32], stride1[15:0]
            g1.s7 = 0;
            asm volatile("tensor_load_to_lds %0, %1"
                         :: "s"(g0), "s"(g1) : "memory");
            asm volatile("s_wait_tensorcnt 0" ::: "memory");
        }
    } else if (UNI_MODE == 1) {
        // Per-lane async DMA (b128 per lane) of the pre-fused table into LDS.
        const unsigned nvec = UNI_PAD / 4u;
        const unsigned lds_base = (unsigned)(unsigned long long)&s_uni[0];
        for (unsigned v = tid; v < nvec; v += THREADS) {
            unsigned boff = v * 16u;
            unsigned lds_addr = lds_base + boff;
            asm volatile("global_load_async_to_lds_b128 %0, %1, %2"
                         :: "v"(lds_addr), "v"(boff), "s"(ws_uni)
                         : "memory");
        }
        asm volatile("s_wait_asynccnt 0" ::: "memory");
    } else {
        // Streaming (coalesced) in-kernel fuse of the three unigram tables.
        for (unsigned i = tid; i < UNI_PAD; i += THREADS) {
            float v = 0.0f;
            if (i < VOCAB && ign[i] <= 0)
                v = (has_uni[i] > 0 ? uni[i] : 0.5f) + 1.0f;
            s_uni[i] = v;
        }
    }
    __syncthreads();

    for (int r = 0; r < ROWS_PER_BLOCK; ++r) {
        int row = blockIdx.x * ROWS_PER_BLOCK + r;
        if (row >= rows) break;                 // uniform across block

        if (tid == 0) s_rowsum = 0.0f;
        __syncthreads();

        float acc = 0.0f;
        const int* __restrict__ xr = x + (size_t)row * (size_t)cols;

        for (int base = (int)tid * 4; base < cols; base += THREADS * 4) {
            int4 c = *(const int4*)(xr + base);          // coalesced 16B/lane
            const int  nb       = base + 4;
            const bool has_next = nb < cols;
            const int  nxt      = has_next ? xr[nb] : 0;

            float f0 = s_uni[c.x];
            float f1 = s_uni[c.y];
            float f2 = s_uni[c.z];
            float f3 = s_uni[c.w];

            acc += (f0 > 0.0f ? f0 - 1.0f : 0.0f);
            acc += (f1 > 0.0f ? f1 - 1.0f : 0.0f);
            acc += (f2 > 0.0f ? f2 - 1.0f : 0.0f);
            acc += (f3 > 0.0f ? f3 - 1.0f : 0.0f);

            acc += bigram_val<BI_MODE>(c.x, c.y, f0, f1, bi, has_bi, ws_bi);
            acc += bigram_val<BI_MODE>(c.y, c.z, f1, f2, bi, has_bi, ws_bi);
            acc += bigram_val<BI_MODE>(c.z, c.w, f2, f3, bi, has_bi, ws_bi);
            if (has_next) {
                float fn = s_uni[nxt];
                acc += bigram_val<BI_MODE>(c.w, nxt, f3, fn, bi, has_bi, ws_bi);
            }
        }

        // wave32 reduction, then one ds_add_f32 per wave
        for (int off = 16; off > 0; off >>= 1)
            acc += __shfl_down(acc, off, 32);
        if ((tid & 31u) == 0)
            atomicAdd(&s_rowsum, acc);
        __syncthreads();
        if (tid == 0) out[row] = s_rowsum;
        // tid 0's out-read precedes its next-iteration zero-write (program order);
        // other threads can't touch s_rowsum until the next barrier -> safe
    }
}

// ---------------- launch ----------------
extern "C" void kernel_launch(void* const* d_in, const int* in_sizes, int n_in,
                              void* d_out, int out_size, void* d_ws, size_t ws_size,
                              hipStream_t stream) {
    const int*   x       = (const int*)  d_in[0];
    const float* uni     = (const float*)d_in[1];
    const int*   has_uni = (const int*)  d_in[2];
    const float* bi      = (const float*)d_in[3];
    const int*   has_bi  = (const int*)  d_in[4];
    const int*   ign     = (const int*)  d_in[5];
    float*       out     = (float*)      d_out;

    const int rows = out_size;                             // 4096
    const int cols = (rows > 0) ? in_sizes[0] / rows : 0;  // 2048

    const size_t needA = (size_t)UNI_PAD * sizeof(float);
    const size_t needB = needA + (size_t)BI_HASH * sizeof(float);
    float* ws_uni = (float*)d_ws;
    float* ws_bi  = (float*)((char*)d_ws + needA);

    const int grid = (rows + ROWS_PER_BLOCK - 1) / ROWS_PER_BLOCK;

    if (d_ws && ws_size >= needB) {
        tox_prep_uni<<<(UNI_PAD + 255) / 256, 256, 0, stream>>>(uni, has_uni, ign, ws_uni);
        tox_prep_bi <<<(BI_HASH + 255) / 256, 256, 0, stream>>>(bi, has_bi, ws_bi);
        tox_main<2, 1><<<grid, THREADS, 0, stream>>>(x, rows, cols, uni, has_uni, ign,
                                                     bi, has_bi, ws_uni, ws_bi, out);
    } else if (d_ws && ws_size >= needA) {
        tox_prep_uni<<<(UNI_PAD + 255) / 256, 256, 0, stream>>>(uni, has_uni, ign, ws_uni);
        tox_main<1, 0><<<grid, THREADS, 0, stream>>>(x, rows, cols, uni, has_uni, ign,
                                                     bi, has_bi, ws_uni, nullptr, out);
    } else {
        tox_main<0, 0><<<grid, THREADS, 0, stream>>>(x, rows, cols, uni, has_uni, ign,
                                                     bi, has_bi, nullptr, nullptr, out);
    }
}